// SpatialEmbLoss_86672440033622
// MI455X (gfx1250) — compile-verified
//
#include <hip/hip_runtime.h>
#include <hip/hip_bf16.h>
#include <math.h>

#define B_ 4
#define H_ 1024
#define W_ 1024
#define HW_ (H_ * W_)
#define NINST 8
#define NSEG (B_ * NINST)
#define NBINS 65536
#define HIST_BYTES ((size_t)NSEG * NBINS * 2 * sizeof(unsigned))

typedef __attribute__((ext_vector_type(2))) float v2f;
typedef __attribute__((ext_vector_type(8))) float v8f;

__device__ __forceinline__ float jaccf(float T, float P, float gts) {
    if (T <= 0.0f) return 0.0f;          // matches reference concat([jacc[:1], diff]) semantics
    float inter = gts - P;
    float uni   = gts + (T - P);
    return 1.0f - inter / fmaxf(uni, 1.0f);
}

// ---------------------------------------------------------------------------
// Kernel 1: per-instance stats (count, sum x, sum y, sum sigma_x, sum sigma_y)
// as a one-hot-mask GEMM using V_WMMA_F32_16X16X4_F32.
//   A (16x4): rows = instance slots (inst id = row+1), K = 4 pixels
//   B (4x16): rows = 4 pixels, cols = features [1, x, y, sx, sy, 0...]
//   D (16x16) accumulates sum_pixels mask * feature.
// Feature fetch is branchless: per-lane loop-invariant base pointer + fma
// fixup, so the hot loop is 2x global_load_b64 + VALU + WMMA, no exec-mask
// divergence.
// ---------------------------------------------------------------------------
__global__ void stats_wmma_kernel(const float* __restrict__ pred,
                                  const int* __restrict__ inst,
                                  const float* __restrict__ coords,
                                  float* __restrict__ stats /* [B][8][8] */) {
    const int b    = blockIdx.y;
    const int lane = threadIdx.x & 31;
    const int grp  = lane >> 4;   // 0 -> K=0,1 ; 1 -> K=2,3
    const int idx  = lane & 15;   // A row (M) and B col (N)

    const int wavesPerBatch = gridDim.x * (blockDim.x >> 5);
    const int gwave = blockIdx.x * (blockDim.x >> 5) + (threadIdx.x >> 5);

    const int*   ib = inst + (size_t)b * HW_;
    const float* sx = pred + ((size_t)b * 5 + 2) * HW_;
    const float* sy = pred + ((size_t)b * 5 + 3) * HW_;
    const float* cx = coords;
    const float* cy = coords + HW_;

    // Loop-invariant per-lane feature source (selected once; no divergence in loop).
    const float* fbase = cx;                 // dummy for idx==0 and idx>=5
    if (idx == 2) fbase = cy;
    else if (idx == 3) fbase = sx;
    else if (idx == 4) fbase = sy;
    const float fmul = (idx >= 1 && idx <= 4) ? 1.0f : 0.0f;
    const float fadd = (idx == 0) ? 1.0f : 0.0f;

    // Each wave owns a contiguous span of pixels.
    const int span  = HW_ / wavesPerBatch;   // grid chosen so this divides exactly
    const int start = gwave * span;

    v8f c = {};
#pragma unroll 4
    for (int p = start; p < start + span; p += 4) {
        const int p0 = p + grp * 2;
        const int p1 = p0 + 1;
        __builtin_prefetch(&fbase[p0 + 64], 0, 1);   // gfx1250 global_prefetch_b8

        const int   i0 = ib[p0];
        const int   i1 = ib[p1];
        const float f0 = fbase[p0] * fmul + fadd;
        const float f1 = fbase[p1] * fmul + fadd;

        v2f a, bb;
        a[0]  = (i0 == idx + 1) ? 1.0f : 0.0f;   // rows 8..15 auto-zero (inst <= 8)
        a[1]  = (i1 == idx + 1) ? 1.0f : 0.0f;
        bb[0] = f0;
        bb[1] = f1;
        // 8 args: (neg_a, A, neg_b, B, c_mod, C, reuse_a, reuse_b)
        c = __builtin_amdgcn_wmma_f32_16x16x4_f32(false, a, false, bb,
                                                  (short)0, c, false, false);
    }
    // C/D layout: lanes 0-15 -> N=lane, M=vgpr v (0..7). Rows 8..15 (lanes 16-31) are zero.
    if (lane < 16 && idx < 5) {
#pragma unroll
        for (int v = 0; v < 8; ++v) {
            atomicAdd(&stats[(size_t)b * 64 + v * 8 + idx], c[v]);
        }
    }
}

// ---------------------------------------------------------------------------
// Kernel 2: finalize per-(b,n) parameters.
// params[seg*8 + {0..7}] = cx, cy, smx, smy, s1=exp(10*smx), s2=exp(10*smy), nclamp, count
// ---------------------------------------------------------------------------
__global__ void finalize_stats_kernel(const float* __restrict__ stats,
                                      float* __restrict__ params) {
    const int seg = threadIdx.x;
    if (seg >= NSEG) return;
    const int b = seg / NINST, n = seg % NINST;
    const float* s = stats + (size_t)b * 64 + n * 8;
    const float cnt = s[0];
    const float ncl = fmaxf(cnt, 1.0f);
    const float smx = s[3] / ncl, smy = s[4] / ncl;
    float* pr = params + seg * 8;
    pr[0] = s[1] / ncl;  pr[1] = s[2] / ncl;
    pr[2] = smx;         pr[3] = smy;
    pr[4] = __expf(10.0f * smx);
    pr[5] = __expf(10.0f * smy);
    pr[6] = ncl;         pr[7] = cnt;
}

// ---------------------------------------------------------------------------
// Kernel 3: streaming per-pixel pass — dist/error histogram, var, seed losses.
// ---------------------------------------------------------------------------
__global__ void pixel_pass_kernel(const float* __restrict__ pred,
                                  const int* __restrict__ inst,
                                  const float* __restrict__ coords,
                                  const float* __restrict__ params,
                                  unsigned* __restrict__ hist,
                                  float* __restrict__ varAcc,
                                  float* __restrict__ fgAcc,
                                  float* __restrict__ bgAcc) {
    const int b = blockIdx.y;
    const int p = blockIdx.x * blockDim.x + threadIdx.x;

    __shared__ float sVar[NINST];
    __shared__ float sFg, sBg;
    if (threadIdx.x < NINST) sVar[threadIdx.x] = 0.0f;
    if (threadIdx.x == 0) { sFg = 0.0f; sBg = 0.0f; }
    __syncthreads();

    const float px = coords[p];
    const float py = coords[HW_ + p];
    const size_t bb = (size_t)b * 5 * HW_;
    const float p0  = pred[bb + p];
    const float p1  = pred[bb + (size_t)HW_ + p];
    const float sgx = pred[bb + (size_t)2 * HW_ + p];
    const float sgy = pred[bb + (size_t)3 * HW_ + p];
    const float p4  = pred[bb + (size_t)4 * HW_ + p];
    const int   iv  = inst[(size_t)b * HW_ + p];

    const float ex   = tanhf(p0) + px;
    const float ey   = tanhf(p1) + py;
    const float seed = 1.0f / (1.0f + __expf(-p4));

    const float binScale = (float)NBINS / 2.002f;
#pragma unroll
    for (int n = 0; n < NINST; ++n) {
        const float* pr = params + (b * NINST + n) * 8;
        const float dx = ex - pr[0];
        const float dy = ey - pr[1];
        const float dist = __expf(-(dx * dx * pr[4] + dy * dy * pr[5]));
        const int lab = (iv == n + 1) ? 1 : 0;
        float e = lab ? (2.0f - 2.0f * dist) : (2.0f * dist);
        e = fmaxf(e, 0.0f);
        int bin = (int)(e * binScale);
        bin = min(bin, NBINS - 1);
        atomicAdd(&hist[(((size_t)(b * NINST + n)) * NBINS + (size_t)bin) * 2 + lab], 1u);
        if (lab) {
            const float vx = sgx - pr[2];
            const float vy = sgy - pr[3];
            atomicAdd(&sVar[n], vx * vx + vy * vy);
            const float fd = seed - dist;
            atomicAdd(&sFg, fd * fd);
        }
    }
    if (iv == 0) atomicAdd(&sBg, seed * seed);
    __syncthreads();
    if (threadIdx.x < NINST) atomicAdd(&varAcc[b * NINST + threadIdx.x], sVar[threadIdx.x]);
    if (threadIdx.x == 0) { atomicAdd(&fgAcc[b], sFg); atomicAdd(&bgAcc[b], sBg); }
}

// ---------------------------------------------------------------------------
// Kernel 4: Lovasz via descending histogram scan (tie-invariant telescoping).
// One workgroup per (b,n) segment.
// ---------------------------------------------------------------------------
__global__ void lovasz_scan_kernel(const unsigned* __restrict__ hist,
                                   const float* __restrict__ params,
                                   float* __restrict__ lov) {
    const int seg = blockIdx.x;
    const int t   = threadIdx.x;
    const float gts = params[seg * 8 + 7];
    const unsigned* h = hist + (size_t)seg * NBINS * 2;

    __shared__ float sT[256], sP[256];
    __shared__ float carry[2];
    if (t == 0) { carry[0] = 0.0f; carry[1] = 0.0f; }

    float acc = 0.0f;
    const float invScale = 2.002f / (float)NBINS;

    for (int chunk = 0; chunk < NBINS / 256; ++chunk) {
        const int bin = NBINS - 1 - (chunk * 256 + t);
        const float cn  = (float)h[(size_t)bin * 2 + 0];
        const float cp  = (float)h[(size_t)bin * 2 + 1];
        const float tot = cn + cp;

        __syncthreads();
        sT[t] = tot; sP[t] = cp;
        __syncthreads();
        // Hillis-Steele inclusive scan over the 256-bin chunk (descending order)
        for (int off = 1; off < 256; off <<= 1) {
            const float aT = (t >= off) ? sT[t - off] : 0.0f;
            const float aP = (t >= off) ? sP[t - off] : 0.0f;
            __syncthreads();
            sT[t] += aT; sP[t] += aP;
            __syncthreads();
        }
        const float cT = carry[0], cP = carry[1];
        const float Tafter  = cT + sT[t];
        const float Pafter  = cP + sP[t];
        const float Tbefore = Tafter - tot;
        const float Pbefore = Pafter - cp;
        if (tot > 0.0f) {
            const float eRep = ((float)bin + 0.5f) * invScale;   // relu(e) == e (e >= 0)
            acc += eRep * (jaccf(Tafter, Pafter, gts) - jaccf(Tbefore, Pbefore, gts));
        }
        __syncthreads();
        if (t == 255) { carry[0] = cT + sT[255]; carry[1] = cP + sP[255]; }
    }

    __syncthreads();
    sT[t] = acc;
    __syncthreads();
    for (int off = 128; off > 0; off >>= 1) {
        if (t < off) sT[t] += sT[t + off];
        __syncthreads();
    }
    if (t == 0) lov[seg] = sT[0];
}

// ---------------------------------------------------------------------------
// Kernel 5: final scalar reduction.
// ---------------------------------------------------------------------------
__global__ void final_kernel(const float* __restrict__ params,
                             const float* __restrict__ lov,
                             const float* __restrict__ varAcc,
                             const float* __restrict__ fgAcc,
                             const float* __restrict__ bgAcc,
                             float* __restrict__ out) {
    if (threadIdx.x != 0 || blockIdx.x != 0) return;
    float loss = 0.0f;
    for (int b = 0; b < B_; ++b) {
        float il = 0.0f, vl = 0.0f;
        for (int n = 0; n < NINST; ++n) {
            const int seg = b * NINST + n;
            il += lov[seg];
            vl += varAcc[seg] / (2.0f * params[seg * 8 + 6]);
        }
        il *= (1.0f / NINST);
        vl *= (1.0f / NINST);
        const float sl = (bgAcc[b] + fgAcc[b]) / (float)HW_;
        loss += il + vl + sl;
    }
    out[0] = loss / (float)B_;
}

extern "C" void kernel_launch(void* const* d_in, const int* in_sizes, int n_in,
                              void* d_out, int out_size, void* d_ws, size_t ws_size,
                              hipStream_t stream) {
    const float* pred   = (const float*)d_in[0];
    const int*   inst   = (const int*)d_in[1];
    const float* coords = (const float*)d_in[2];

    unsigned* hist = (unsigned*)d_ws;
    float* w      = (float*)((char*)d_ws + HIST_BYTES);
    float* stats  = w;          // 256 floats: [B][8][8]
    float* params = w + 256;    // 256 floats: [NSEG][8]
    float* varAcc = w + 512;    // 32
    float* fgAcc  = w + 544;    // 4
    float* bgAcc  = w + 548;    // 4
    float* lov    = w + 552;    // 32

    hipMemsetAsync(d_ws, 0, HIST_BYTES + 4096, stream);

    // 512 blocks x 8 waves = 4096 waves per batch -> each wave owns a
    // contiguous 256-pixel span (64 iterations of 4-pixel WMMAs).
    stats_wmma_kernel<<<dim3(512, B_), 256, 0, stream>>>(pred, inst, coords, stats);
    finalize_stats_kernel<<<1, 64, 0, stream>>>(stats, params);
    pixel_pass_kernel<<<dim3(HW_ / 256, B_), 256, 0, stream>>>(pred, inst, coords,
                                                               params, hist,
                                                               varAcc, fgAcc, bgAcc);
    lovasz_scan_kernel<<<NSEG, 256, 0, stream>>>(hist, params, lov);
    final_kernel<<<1, 32, 0, stream>>>(params, lov, varAcc, fgAcc, bgAcc, (float*)d_out);
}